// MemoryModel_66159676228023
// MI455X (gfx1250) — compile-verified
//
#include <hip/hip_runtime.h>
#include <hip/hip_bf16.h>
#include <math.h>

typedef __attribute__((ext_vector_type(2))) float v2f;
typedef __attribute__((ext_vector_type(8))) float v8f;

#define REG    0.1f
#define REG2   0.01f
#define N_ROWS 1024
#define HDIM   128
#define NHTOT  (N_ROWS * HDIM)
#define KPAD   176   // DIN(172) + 2*ED(2) padded 174 -> 176 (multiple of 4)
#define SPLITK 4

// ---------------------------------------------------------------------------
// f32 WMMA GEMM, V_WMMA_F32_16X16X4_F32, 32x32 output tile per wave (2x2
// register tiling: 2 A frags + 2 B frags -> 4 WMMAs per K-step, halving the
// load-instruction count per WMMA vs a 16x16 tile). blockIdx.y = K-split
// index; each split writes its own M*N slice (summed by reduce_splits_kernel
// in a fixed order => deterministic).
// Frag layouts (ISA 7.12.2):
//   A 16x4 : lane<16 -> M=lane, K=k..k+1 ; lane>=16 -> M=lane-16, K=k+2..k+3
//   B 4x16 : lane<16 -> N=lane, K rows k,k+1 ; lane>=16 -> K rows k+2,k+3
//   D 16x16: vgpr v: lanes<16 -> M=v, lanes>=16 -> M=8+v ; N=lane%16
// ---------------------------------------------------------------------------
__global__ __launch_bounds__(256)
void gemm_f32_wmma(const float* __restrict__ A, const float* __restrict__ B,
                   float* __restrict__ C, int M, int N, int K, int kChunk) {
    const int lane = threadIdx.x & 31;
    const int wave = (blockIdx.x * blockDim.x + threadIdx.x) >> 5;
    const int tilesN2 = N >> 5;                 // 32-wide column groups
    const int ti = wave / tilesN2;              // 32-row strip
    const int tj = wave - ti * tilesN2;
    if (ti * 32 >= M) return;

    const int s  = blockIdx.y;
    const int k0 = s * kChunk;
    int k1 = k0 + kChunk; if (k1 > K) k1 = K;

    const int half = lane >> 4;                 // selects K pair within frag
    const int r    = lane & 15;

    const float* __restrict__ A0 = A + (size_t)(ti * 32 + r) * K + 2 * half;
    const float* __restrict__ A1 = A0 + (size_t)16 * K;
    const float* __restrict__ B0 = B + (size_t)(2 * half) * N + tj * 32 + r;
    const float* __restrict__ B1 = B0 + 16;

    v8f acc00 = {}, acc01 = {}, acc10 = {}, acc11 = {};
#pragma unroll 2
    for (int k = k0; k < k1; k += 4) {
        v2f a0, a1, b0, b1;
        a0.x = A0[k];               a0.y = A0[k + 1];
        a1.x = A1[k];               a1.y = A1[k + 1];
        b0.x = B0[(size_t)k * N];   b0.y = B0[(size_t)(k + 1) * N];
        b1.x = B1[(size_t)k * N];   b1.y = B1[(size_t)(k + 1) * N];
        acc00 = __builtin_amdgcn_wmma_f32_16x16x4_f32(false, a0, false, b0, (short)0, acc00, false, false);
        acc01 = __builtin_amdgcn_wmma_f32_16x16x4_f32(false, a0, false, b1, (short)0, acc01, false, false);
        acc10 = __builtin_amdgcn_wmma_f32_16x16x4_f32(false, a1, false, b0, (short)0, acc10, false, false);
        acc11 = __builtin_amdgcn_wmma_f32_16x16x4_f32(false, a1, false, b1, (short)0, acc11, false, false);
    }

    float* __restrict__ C0 = C + (size_t)s * M * N
                               + (size_t)(ti * 32 + 8 * half) * N + tj * 32 + r;
#pragma unroll
    for (int v = 0; v < 8; ++v) {
        C0[(size_t)v * N]              = acc00[v];
        C0[(size_t)v * N + 16]         = acc01[v];
        C0[(size_t)(v + 16) * N]       = acc10[v];
        C0[(size_t)(v + 16) * N + 16]  = acc11[v];
    }
}

// C[idx] = sum over split slices (fixed order -> deterministic)
__global__ __launch_bounds__(256)
void reduce_splits_kernel(const float* __restrict__ P, float* __restrict__ C, int n) {
    int idx = blockIdx.x * blockDim.x + threadIdx.x;
    if (idx >= n) return;
    float s = 0.0f;
#pragma unroll
    for (int j = 0; j < SPLITK; ++j) s += P[(size_t)j * n + idx];
    C[idx] = s;
}

// ---------------------------------------------------------------------------
// Pad W_tune [174,128] -> [176,128] (zero rows 174..175)
// ---------------------------------------------------------------------------
__global__ void pad_wtune_kernel(const float* __restrict__ W, float* __restrict__ Wp) {
    int idx = blockIdx.x * blockDim.x + threadIdx.x;
    if (idx >= KPAD * HDIM) return;
    int row = idx / HDIM, col = idx - row * HDIM;
    Wp[idx] = (row < 174) ? W[row * HDIM + col] : 0.0f;
}

// ---------------------------------------------------------------------------
// Build x_in [1024,176] = [x_sub | neigh(2) | pad(2)]
// neigh rows 0..255:   (names[src[e]], names[dst[e]])
// neigh rows 256..511: (names[dst[e]], names[src[e]])   (ED==1)
// ---------------------------------------------------------------------------
__global__ __launch_bounds__(128)
void build_xin_kernel(const float* __restrict__ xs, const float* __restrict__ names,
                      const int* __restrict__ src, const int* __restrict__ dst,
                      float* __restrict__ xin) {
    int i = blockIdx.x;
    for (int c = threadIdx.x; c < KPAD; c += blockDim.x) {
        float v = 0.0f;
        if (c < 172) {
            v = xs[i * 172 + c];
        } else if (c < 174) {
            int col = c - 172;
            if (i < 256) {
                int e = i;
                v = (col == 0) ? names[src[e]] : names[dst[e]];
            } else if (i < 512) {
                int e = i - 256;
                v = (col == 0) ? names[dst[e]] : names[src[e]];
            }
        }
        xin[i * KPAD + c] = v;
    }
}

// ---------------------------------------------------------------------------
// Per-row: zn = rmsnorm(z, scale); delta = softplus(zn @ W_dt + b_dt)
// ---------------------------------------------------------------------------
__global__ __launch_bounds__(128)
void rms_delta_kernel(const float* __restrict__ z, const float* __restrict__ scale,
                      const float* __restrict__ Wdt, const float* __restrict__ bdt,
                      float* __restrict__ zn, float* __restrict__ delta) {
    const int i = blockIdx.x, h = threadIdx.x;
    __shared__ float red[HDIM];
    float x = z[i * HDIM + h];
    red[h] = x * x;
    __syncthreads();
    for (int s = 64; s > 0; s >>= 1) { if (h < s) red[h] += red[h + s]; __syncthreads(); }
    float rms = sqrtf(red[0]) * 0.08838834764831845f;   // 1/sqrt(128)
    float znv = scale[h] * x / (rms + 1e-8f);
    zn[i * HDIM + h] = znv;
    __syncthreads();
    red[h] = znv * Wdt[h];
    __syncthreads();
    for (int s = 64; s > 0; s >>= 1) { if (h < s) red[h] += red[h + s]; __syncthreads(); }
    if (h == 0) {
        float v = red[0] + bdt[0];
        delta[i] = (v > 0.0f) ? (v + log1pf(expf(-v))) : log1pf(expf(v));
    }
}

// out = (in + bias[h]) * rowscale[i]   (rowscale==nullptr -> 1)
__global__ __launch_bounds__(256)
void bias_rowscale_kernel(const float* __restrict__ in, const float* __restrict__ bias,
                          const float* __restrict__ rowscale, float* __restrict__ out, int n) {
    int idx = blockIdx.x * blockDim.x + threadIdx.x;
    if (idx >= n) return;
    int h = idx & (HDIM - 1);
    int i = idx >> 7;
    float s = rowscale ? rowscale[i] : 1.0f;
    out[idx] = (in[idx] + bias[h]) * s;
}

// out = xin - REG * g     (P_L_inv @ .)
__global__ __launch_bounds__(256)
void plinv_kernel(const float* __restrict__ xin, const float* __restrict__ g,
                  float* __restrict__ out, int n) {
    int idx = blockIdx.x * blockDim.x + threadIdx.x;
    if (idx >= n) return;
    out[idx] = xin[idx] - REG * g[idx];
}

// V[i,h] = m[ids[i],h] * exp(delta[i] * -exp(Alog[h]))
__global__ __launch_bounds__(256)
void gather_atbar_kernel(const float* __restrict__ mvec, const int* __restrict__ ids,
                         const float* __restrict__ delta, const float* __restrict__ Alog,
                         float* __restrict__ V, int n) {
    int idx = blockIdx.x * blockDim.x + threadIdx.x;
    if (idx >= n) return;
    int i = idx >> 7, h = idx & (HDIM - 1);
    float dA = delta[i] * (-expf(Alog[h]));
    V[idx] = mvec[(size_t)ids[i] * HDIM + h] * expf(dA);
}

// ---------------------------------------------------------------------------
// GL quadrature + As_bar@V combine + (optional) u2 = zt + gelu_tanh(uc)
// integral = sum_k w_k * exp(dA*t_k) * (X - R*t_k*Y1 + R^2*t_k*Y2 + R^2*t_k^2/2*Y3)
// uc = V - R*Rb + R^2*Qb + R^2/2*Sb + integral
// ---------------------------------------------------------------------------
__global__ __launch_bounds__(256)
void finale_kernel(const float* __restrict__ X,  const float* __restrict__ Y1,
                   const float* __restrict__ Y2, const float* __restrict__ Y3,
                   const float* __restrict__ V,  const float* __restrict__ Rb,
                   const float* __restrict__ Qb, const float* __restrict__ Sb,
                   const float* __restrict__ delta, const float* __restrict__ Alog,
                   const float* __restrict__ zt, float* __restrict__ out_uc,
                   float* __restrict__ u2, int do_u2, int n) {
    int idx = blockIdx.x * blockDim.x + threadIdx.x;
    if (idx >= n) return;
    const int i = idx >> 7, h = idx & (HDIM - 1);

    const float nodes[8] = {-0.1834346424956498f, -0.525532409916329f,
                            -0.7966664774136267f, -0.9602898564975363f,
                             0.1834346424956498f,  0.525532409916329f,
                             0.7966664774136267f,  0.9602898564975363f};
    const float glw[8]   = {0.362683783378362f, 0.3137066458778873f,
                            0.2223810344533745f, 0.1012285362903763f,
                            0.362683783378362f, 0.3137066458778873f,
                            0.2223810344533745f, 0.1012285362903763f};

    float dA = delta[i] * (-expf(Alog[h]));
    float x = X[idx], y1 = Y1[idx], y2 = Y2[idx], y3 = Y3[idx];
    float integral = 0.0f;
#pragma unroll
    for (int k = 0; k < 8; ++k) {
        float t = 0.5f * (nodes[k] + 1.0f);
        float w = 0.5f * glw[k];
        float lhs = x - REG * t * y1 + REG2 * t * y2 + 0.5f * REG2 * t * t * y3;
        integral += w * expf(dA * t) * lhs;
    }
    float uc = V[idx] - REG * Rb[idx] + REG2 * Qb[idx] + 0.5f * REG2 * Sb[idx] + integral;
    out_uc[idx] = uc;
    if (do_u2) {
        float g = 0.5f * uc * (1.0f + tanhf(0.7978845608028654f * (uc + 0.044715f * uc * uc * uc)));
        u2[idx] = zt[idx] + g;
    }
}

// ---------------------------------------------------------------------------
extern "C" void kernel_launch(void* const* d_in, const int* in_sizes, int n_in,
                              void* d_out, int out_size, void* d_ws, size_t ws_size,
                              hipStream_t stream) {
    const float* L    = (const float*)d_in[0];
    const float* dL   = (const float*)d_in[1];
    const float* xs   = (const float*)d_in[2];
    const float* m1   = (const float*)d_in[3];
    const float* m2   = (const float*)d_in[4];
    const float* names= (const float*)d_in[5];
    const float* rms1 = (const float*)d_in[6];
    const float* rms2 = (const float*)d_in[7];
    const float* Wt   = (const float*)d_in[8];
    const float* bt   = (const float*)d_in[9];
    const float* WB1  = (const float*)d_in[10];
    const float* bB1  = (const float*)d_in[11];
    const float* WB2  = (const float*)d_in[12];
    const float* bB2  = (const float*)d_in[13];
    const float* Wdt  = (const float*)d_in[14];
    const float* bdt  = (const float*)d_in[15];
    const float* Al1  = (const float*)d_in[16];
    const float* Al2  = (const float*)d_in[17];
    const int*   src  = (const int*)d_in[18];
    const int*   dst  = (const int*)d_in[19];
    const int*   ids  = (const int*)d_in[20];

    float* w = (float*)d_ws;
    float* xin  = w; w += N_ROWS * KPAD;
    float* wpad = w; w += KPAD * HDIM;
    float* gpart= w; w += SPLITK * NHTOT;   // split-K partial sums
    float* zt   = w; w += NHTOT;
    float* zn   = w; w += NHTOT;
    float* raw  = w; w += NHTOT;
    float* xsb  = w; w += NHTOT;            // (B*z + b) * delta
    float* X    = w; w += NHTOT;
    float* Tb   = w; w += NHTOT;
    float* Y1   = w; w += NHTOT;
    float* Y2   = w; w += NHTOT;
    float* Y3   = w; w += NHTOT;
    float* V    = w; w += NHTOT;
    float* Pb   = w; w += NHTOT;
    float* Rb   = w; w += NHTOT;
    float* Qb   = w; w += NHTOT;
    float* Sb   = w; w += NHTOT;
    float* u2   = w; w += NHTOT;
    float* delta= w; w += N_ROWS;

    const int EW_BLOCKS = NHTOT / 256;

    auto gemm = [&](const float* A, const float* B, float* C, int M, int N, int K) {
        int splits = (K >= 512) ? SPLITK : 1;
        int kChunk = ((K + splits - 1) / splits + 3) & ~3;
        int waves  = (M / 32) * (N / 32);
        dim3 grid((waves * 32 + 255) / 256, splits);
        float* dst = (splits == 1) ? C : gpart;
        gemm_f32_wmma<<<grid, 256, 0, stream>>>(A, B, dst, M, N, K, kChunk);
        if (splits > 1)
            reduce_splits_kernel<<<(M * N + 255) / 256, 256, 0, stream>>>(gpart, C, M * N);
    };

    // ---- preamble: zt = [x_sub | neigh] @ W_tune + b_tune -------------------
    pad_wtune_kernel<<<(KPAD * HDIM + 255) / 256, 256, 0, stream>>>(Wt, wpad);
    build_xin_kernel<<<N_ROWS, 128, 0, stream>>>(xs, names, src, dst, xin);
    gemm(xin, wpad, raw, N_ROWS, HDIM, KPAD);
    bias_rowscale_kernel<<<EW_BLOCKS, 256, 0, stream>>>(raw, bt, nullptr, zt, NHTOT);

    // ---- one SSM stage ------------------------------------------------------
    auto stage = [&](const float* scale, const float* WB, const float* bB,
                     const float* mv, const float* Al, const float* inz,
                     float* out_uc, float* u2out, int do_u2) {
        rms_delta_kernel<<<N_ROWS, 128, 0, stream>>>(inz, scale, Wdt, bdt, zn, delta);
        gemm(zn, WB, raw, N_ROWS, HDIM, HDIM);
        bias_rowscale_kernel<<<EW_BLOCKS, 256, 0, stream>>>(raw, bB, delta, xsb, NHTOT);
        // X = P_L_inv @ xsb = xsb - REG * (L @ xsb)
        gemm(L, xsb, Tb, N_ROWS, HDIM, N_ROWS);
        plinv_kernel<<<EW_BLOCKS, 256, 0, stream>>>(xsb, Tb, X, NHTOT);
        // LHS_k @ X composition terms
        gemm(dL, X,  Y1, N_ROWS, HDIM, N_ROWS);   // dL @ X
        gemm(L,  X,  Tb, N_ROWS, HDIM, N_ROWS);   // L @ X
        gemm(dL, Tb, Y2, N_ROWS, HDIM, N_ROWS);   // dL @ (L @ X)
        gemm(dL, Y1, Y3, N_ROWS, HDIM, N_ROWS);   // dL @ (dL @ X)
        // V = m[ids] * At_bar ; As_bar @ V composition terms
        gather_atbar_kernel<<<EW_BLOCKS, 256, 0, stream>>>(mv, ids, delta, Al, V, NHTOT);
        gemm(L,  V,  Pb, N_ROWS, HDIM, N_ROWS);   // L @ V
        gemm(dL, V,  Rb, N_ROWS, HDIM, N_ROWS);   // dL @ V
        gemm(dL, Pb, Qb, N_ROWS, HDIM, N_ROWS);   // dL @ (L @ V)
        gemm(dL, Rb, Sb, N_ROWS, HDIM, N_ROWS);   // dL @ (dL @ V)
        finale_kernel<<<EW_BLOCKS, 256, 0, stream>>>(X, Y1, Y2, Y3, V, Rb, Qb, Sb,
                                                     delta, Al, zt, out_uc, u2out,
                                                     do_u2, NHTOT);
    };

    float* out = (float*)d_out;
    stage(rms1, WB1, bB1, m1, Al1, zt, out,         u2, 1);   // -> updated_c_1, u_2
    stage(rms2, WB2, bB2, m2, Al2, u2, out + NHTOT, u2, 0);   // -> updated_c_2
}